// minDepthFlowProjectionModule_47253230191032
// MI455X (gfx1250) — compile-verified
//
#include <hip/hip_runtime.h>

// Problem constants from the reference: B=8, H=720, W=1280, fp32 everywhere.
constexpr int Bn = 8;
constexpr int Hn = 720;
constexpr int Wn = 1280;
constexpr int N  = Hn * Wn;        // 921,600 pixels per batch image
constexpr int TOT = Bn * N;        // 7,372,800 source pixels
constexpr float BIGF = 1e30f;
constexpr int T = 256;             // 8 wave32s per block; Wn % T == 0 (1280 = 5*256)

// ---------------------------------------------------------------------------
// Pass 0: zero the output (harness poisons it), init mind=BIG, cnt=0.
// ---------------------------------------------------------------------------
__global__ void k_init(float* __restrict__ out,
                       int*   __restrict__ mind,
                       float* __restrict__ cnt) {
    int i = blockIdx.x * blockDim.x + threadIdx.x;
    if (i < 2 * TOT) out[i] = 0.0f;
    if (i < TOT) {
        mind[i] = __float_as_int(BIGF);   // positive float: int order == float order
        cnt[i]  = 0.0f;
    }
}

// Shared per-pixel warp-target math. Grid supplies (x,y,b) directly — zero
// integer divides anywhere in the hot kernels.
// Streaming inputs are read with TH=NT so they never evict the 118 MB scatter
// working set (mind/cnt/out) from MI455X's 192 MB L2.
__device__ __forceinline__ bool pixel_targets(int x, int y, int b,
                                              const float* __restrict__ flow,
                                              float& fx, float& fy, int& p,
                                              int& t0, int& t1, int& t2, int& t3) {
    p = y * Wn + x;
    int fbase = b * 2 * N;
    fx = __builtin_nontemporal_load(&flow[fbase + p]);       // (B,2,H,W) ch 0
    fy = __builtin_nontemporal_load(&flow[fbase + N + p]);   //           ch 1
    float x2 = (float)x + fx;
    float y2 = (float)y + fy;
    if (!(x2 >= 0.0f && y2 >= 0.0f &&
          x2 <= (float)(Wn - 1) && y2 <= (float)(Hn - 1)))
        return false;
    // valid => floor(x2) in [0, Wn-1], floor(y2) in [0, Hn-1]
    int xL = (int)floorf(x2);
    int yT = (int)floorf(y2);
    int xR = min(xL + 1, Wn - 1);
    int yB = min(yT + 1, Hn - 1);
    int base = b * N;
    t0 = base + yT * Wn + xL;
    t1 = base + yT * Wn + xR;
    t2 = base + yB * Wn + xL;
    t3 = base + yB * Wn + xR;
    return true;
}

// ---------------------------------------------------------------------------
// Pass 1: scatter atomic-min of the depth bit pattern into 4 bilinear corners.
// Depth is strictly positive => IEEE754 bit order == float order, so integer
// atomic-min is an exact float min and later equality is an exact bit compare.
// The atomics land in L2 (resident set fits in 192 MB), never HBM round-trips.
// ---------------------------------------------------------------------------
__global__ void k_min_scatter(const float* __restrict__ flow,
                              const float* __restrict__ depth,
                              int* __restrict__ mind) {
    int x = blockIdx.x * blockDim.x + threadIdx.x;  // [0, 1280)
    int y = blockIdx.y;                             // [0, 720)
    int b = blockIdx.z;                             // [0, 8)
    float fx, fy; int p, t0, t1, t2, t3;
    if (!pixel_targets(x, y, b, flow, fx, fy, p, t0, t1, t2, t3)) return;
    int di = __float_as_int(__builtin_nontemporal_load(&depth[b * N + p]));
    atomicMin(&mind[t0], di);
    atomicMin(&mind[t1], di);
    atomicMin(&mind[t2], di);
    atomicMin(&mind[t3], di);
}

// ---------------------------------------------------------------------------
// Pass 2: for every (pixel, corner) whose depth equals the winning min, scatter
// add -fx/-fy into d_out (accumulated in place) and bump cnt. Duplicate corner
// indices (clamped edges) intentionally contribute multiple times, exactly as
// the reference's .at[gidx].add does with repeated indices.
// mind reads stay regular-temporal: each cell is re-read by ~4 source pixels.
// ---------------------------------------------------------------------------
__global__ void k_add_scatter(const float* __restrict__ flow,
                              const float* __restrict__ depth,
                              const int* __restrict__ mind,
                              float* __restrict__ out,
                              float* __restrict__ cnt) {
    int x = blockIdx.x * blockDim.x + threadIdx.x;
    int y = blockIdx.y;
    int b = blockIdx.z;
    float fx, fy; int p, t0, t1, t2, t3;
    if (!pixel_targets(x, y, b, flow, fx, fy, p, t0, t1, t2, t3)) return;
    int di = __float_as_int(__builtin_nontemporal_load(&depth[b * N + p]));
    int base  = b * N;       // (b, pixel 0) in mind/cnt space
    int obase = b * 2 * N;   // (b, ch0, pixel 0) in output space
    int tgts[4] = {t0, t1, t2, t3};
#pragma unroll
    for (int k = 0; k < 4; ++k) {
        int t = tgts[k];
        if (mind[t] == di) {
            int q = t - base;                       // pixel index within batch
            atomicAdd(&out[obase + q],     -fx);    // channel 0 accumulator
            atomicAdd(&out[obase + N + q], -fy);    // channel 1 accumulator
            atomicAdd(&cnt[t], 1.0f);
        }
    }
}

// ---------------------------------------------------------------------------
// Pass 3: normalize. Matches reference: where(cnt>0, sum/cnt, 0) — the zero
// branch is already in place from k_init. cnt is last-use (NT read); the final
// output is never re-read on device, so store it non-temporally.
// ---------------------------------------------------------------------------
__global__ void k_normalize(float* __restrict__ out,
                            const float* __restrict__ cnt) {
    int x = blockIdx.x * blockDim.x + threadIdx.x;
    int y = blockIdx.y;
    int b = blockIdx.z;
    int p = y * Wn + x;
    float c = __builtin_nontemporal_load(&cnt[b * N + p]);
    if (c > 0.0f) {
        int obase = b * 2 * N;
        float ox = out[obase + p]     / c;
        float oy = out[obase + N + p] / c;
        __builtin_nontemporal_store(ox, &out[obase + p]);
        __builtin_nontemporal_store(oy, &out[obase + N + p]);
    }
}

// ---------------------------------------------------------------------------
// Launch: 4 grid-synchronized passes on `stream`. Scratch: mind (int, B*N) and
// cnt (float, B*N) in d_ws = ~59 MB. Deterministic: all state re-initialized
// every call.
// ---------------------------------------------------------------------------
extern "C" void kernel_launch(void* const* d_in, const int* in_sizes, int n_in,
                              void* d_out, int out_size, void* d_ws, size_t ws_size,
                              hipStream_t stream) {
    const float* flow  = (const float*)d_in[0];   // (B,2,H,W) fp32
    const float* depth = (const float*)d_in[1];   // (B,1,H,W) fp32
    float* out = (float*)d_out;                   // (B,2,H,W) fp32

    int*   mind = (int*)d_ws;                     // B*N ints
    float* cnt  = (float*)((char*)d_ws + (size_t)TOT * sizeof(int));

    dim3 blk(T);
    dim3 grid_init((2 * TOT + T - 1) / T);        // 1D over 2*B*N
    dim3 grid_pix(Wn / T, Hn, Bn);                // (5, 720, 8) — no divides in-kernel

    k_init       <<<grid_init, blk, 0, stream>>>(out, mind, cnt);
    k_min_scatter<<<grid_pix,  blk, 0, stream>>>(flow, depth, mind);
    k_add_scatter<<<grid_pix,  blk, 0, stream>>>(flow, depth, mind, out, cnt);
    k_normalize  <<<grid_pix,  blk, 0, stream>>>(out, cnt);
}